// MSE_KL_Loss_51101520888567
// MI455X (gfx1250) — compile-verified
//
#include <hip/hip_runtime.h>

// ---- problem constants (match the JAX reference) ----
#define NBINS 100
static constexpr float  LM1F     = -4.6051702f;   // log(0.01)
static constexpr float  LM2F     =  5.9914646f;   // log(400)
static constexpr float  BINSCALE =  9.43695710f;  // NBINS / (LM2 - LM1)
static constexpr double EPSD     =  1e-5;
static constexpr double BETAD    =  0.1;

typedef float v2f  __attribute__((ext_vector_type(2)));
typedef float v4f  __attribute__((ext_vector_type(4)));
typedef float v8f  __attribute__((ext_vector_type(8)));

#define THREADS 256
#define WAVES   (THREADS / 32)
#define MAXBLK  2048

__device__ __forceinline__ void hist_add(unsigned int* h, float x) {
    if (x >= LM1F && x <= LM2F) {
        int b = (int)((x - LM1F) * BINSCALE);
        b = (b < NBINS) ? b : (NBINS - 1);     // right edge inclusive in last bin
        atomicAdd(&h[b], 1u);
    }
}

// Wave32 sum-reduction. Preferred path: V_WMMA_F32_16X16X4_F32 with B = ones.
// A (16x4 f32) layout: VGPR0 lane L<16 -> A[L][0], lane L>=16 -> A[L-16][2];
// VGPR1 (K=1,3) set to 0.  =>  D[m][n] = acc[m] + acc[m+16] for every n.
// Lane's 8 C VGPRs cover M=0..7 (lanes 0-15) or M=8..15 (lanes 16-31), so
// sum8(this lane) + sum8(lane^16) == full 32-lane sum.
__device__ __forceinline__ float wave_sum32(float acc) {
#if defined(__gfx1250__) && __has_builtin(__builtin_amdgcn_wmma_f32_16x16x4_f32)
    v2f A; A[0] = acc;  A[1] = 0.0f;
    v2f B; B[0] = 1.0f; B[1] = 1.0f;
    v8f C = {};
    C = __builtin_amdgcn_wmma_f32_16x16x4_f32(
            /*neg_a=*/false, A, /*neg_b=*/false, B,
            /*c_mod=*/(short)0, C, /*reuse_a=*/false, /*reuse_b=*/false);
    float t = ((C[0] + C[1]) + (C[2] + C[3])) + ((C[4] + C[5]) + (C[6] + C[7]));
    t += __shfl_xor(t, 16);
    return t;
#else
    float t = acc;
    #pragma unroll
    for (int o = 16; o > 0; o >>= 1) t += __shfl_xor(t, o);
    return t;
#endif
}

__global__ void zero_hist_kernel(unsigned int* __restrict__ ghist) {
    if (threadIdx.x < 2 * NBINS) ghist[threadIdx.x] = 0u;
}

// Pass 1: streamed over both arrays once. Per-thread f32 SSE accumulation,
// per-wave LDS histograms, WMMA wave reduction, one partial per block.
__global__ void __launch_bounds__(THREADS)
loss_pass1(const float* __restrict__ pred, const float* __restrict__ act,
           float* __restrict__ partials, unsigned int* __restrict__ ghist,
           int nvec, int rem) {
    __shared__ unsigned int lh[WAVES * 2 * NBINS];   // per-wave privatized hist
    __shared__ float        wsum[WAVES];

    const int tid  = threadIdx.x;
    const int wave = tid >> 5;

    for (int i = tid; i < WAVES * 2 * NBINS; i += THREADS) lh[i] = 0u;
    __syncthreads();

    unsigned int* hp = &lh[wave * 2 * NBINS];        // [0..99]=pred, [100..199]=actual

    const v4f* p4 = (const v4f*)pred;
    const v4f* a4 = (const v4f*)act;

    float acc = 0.0f;
    const int stride = gridDim.x * THREADS;
    for (int i = blockIdx.x * THREADS + tid; i < nvec; i += stride) {
        v4f p = __builtin_nontemporal_load(&p4[i]);  // streamed once: NT hint
        v4f a = __builtin_nontemporal_load(&a4[i]);
        float d;
        d = p[0] - a[0]; acc = fmaf(d, d, acc);
        d = p[1] - a[1]; acc = fmaf(d, d, acc);
        d = p[2] - a[2]; acc = fmaf(d, d, acc);
        d = p[3] - a[3]; acc = fmaf(d, d, acc);
        hist_add(hp,         p[0]); hist_add(hp,         p[1]);
        hist_add(hp,         p[2]); hist_add(hp,         p[3]);
        hist_add(hp + NBINS, a[0]); hist_add(hp + NBINS, a[1]);
        hist_add(hp + NBINS, a[2]); hist_add(hp + NBINS, a[3]);
    }

    // scalar tail (N % 4), handled by block 0 only
    if (blockIdx.x == 0 && tid < rem) {
        const int idx = nvec * 4 + tid;
        float p = pred[idx], a = act[idx];
        float d = p - a; acc = fmaf(d, d, acc);
        hist_add(hp, p); hist_add(hp + NBINS, a);
    }

    // wave reduction (EXEC is all ones here: straight-line, no early returns)
    float t = wave_sum32(acc);
    if ((tid & 31) == 0) wsum[wave] = t;
    __syncthreads();

    if (tid == 0) {
        float s = 0.0f;
        #pragma unroll
        for (int w = 0; w < WAVES; ++w) s += wsum[w];
        partials[blockIdx.x] = s;                    // fixed slot -> deterministic
    }

    // merge per-wave histograms into global (integer atomics: exact, deterministic)
    for (int b = tid; b < 2 * NBINS; b += THREADS) {
        unsigned int s = 0;
        #pragma unroll
        for (int w = 0; w < WAVES; ++w) s += lh[w * 2 * NBINS + b];
        if (s) atomicAdd(&ghist[b], s);
    }
}

// Pass 2: single block. Deterministic double-precision finish (trivial cost).
__global__ void __launch_bounds__(THREADS)
loss_final(const float* __restrict__ partials, int nparts,
           const unsigned int* __restrict__ ghist,
           float* __restrict__ out, double invN) {
    __shared__ double sd[THREADS];
    double s = 0.0;
    for (int i = threadIdx.x; i < nparts; i += THREADS) s += (double)partials[i];
    sd[threadIdx.x] = s;
    __syncthreads();
    for (int o = THREADS / 2; o > 0; o >>= 1) {
        if (threadIdx.x < o) sd[threadIdx.x] += sd[threadIdx.x + o];
        __syncthreads();
    }
    if (threadIdx.x == 0) {
        double mse = sd[0] * invN;

        double sp = 0.0, sa = 0.0;
        for (int i = 0; i < NBINS; ++i) {
            sp += (double)ghist[i]         + EPSD;
            sa += (double)ghist[NBINS + i] + EPSD;
        }
        double kld = 0.0;
        for (int i = 0; i < NBINS; ++i) {
            double p = ((double)ghist[i]         + EPSD) / sp;
            double a = ((double)ghist[NBINS + i] + EPSD) / sa;
            kld += a * (log(a) - log(p));
        }
        kld /= (double)NBINS;

        out[0] = (float)(mse + BETAD * kld);
    }
}

extern "C" void kernel_launch(void* const* d_in, const int* in_sizes, int n_in,
                              void* d_out, int out_size, void* d_ws, size_t ws_size,
                              hipStream_t stream) {
    const float* pred = (const float*)d_in[0];
    const float* act  = (const float*)d_in[1];
    const int N = in_sizes[0];

    const int nvec = N >> 2;
    const int rem  = N & 3;

    // workspace layout: [0,800) global hist (200 u32), [1024, 1024+4*blocks) partials
    unsigned int* ghist    = (unsigned int*)d_ws;
    float*        partials = (float*)((char*)d_ws + 1024);

    int blocks = (nvec + THREADS - 1) / THREADS;
    if (blocks > MAXBLK) blocks = MAXBLK;
    if (blocks < 1)      blocks = 1;

    zero_hist_kernel<<<1, THREADS, 0, stream>>>(ghist);
    loss_pass1<<<blocks, THREADS, 0, stream>>>(pred, act, partials, ghist, nvec, rem);
    loss_final<<<1, THREADS, 0, stream>>>(partials, blocks, ghist, (float*)d_out,
                                          1.0 / (double)N);
}